// VARMABaseline_43989055046205
// MI455X (gfx1250) — compile-verified
//
#include <hip/hip_runtime.h>

typedef __attribute__((ext_vector_type(2))) float v2f;
typedef __attribute__((ext_vector_type(8))) float v8f;

// Problem constants (match reference)
#define Bn 256
#define Sn 2048
#define In 6
#define Hn 64

__device__ inline float dev_tanh(float x) {
#if __has_builtin(__builtin_amdgcn_tanhf)
  return __builtin_amdgcn_tanhf(x);
#else
  return tanhf(x);
#endif
}

// One wave32 handles a tile of 16 consecutive timesteps for one batch row.
// Stage 1: hidden_pre[16x64] = X_aug[16x20] @ W_aug[20x64]   (5 K-steps x 4 N-tiles WMMA f32 16x16x4)
// Stage 2: out[16x5]        = tanh(hidden)[16x64] @ W_all[64x16]  (16 K-steps WMMA, 5 live columns)
__global__ __launch_bounds__(256) void varma_wmma_kernel(
    const float* __restrict__ Y,     const float* __restrict__ E,
    const float* __restrict__ W_ar,  const float* __restrict__ b_ar,
    const float* __restrict__ W_ma,  const float* __restrict__ b_ma,
    const float* __restrict__ W_out, const float* __restrict__ b_out,
    const float* __restrict__ W_gh,  const float* __restrict__ b_gh,
    const float* __restrict__ W_ga,  const float* __restrict__ b_ga,
    float* __restrict__ out)
{
  // Pair-contiguous weight layouts: (k0,k0+1) adjacent -> single aligned ds_load_b64
  __shared__ float WaugP[64][20];      // [h][k] augmented input weights (stride 20: conflict-free)
  __shared__ float Wo2P[16][68];       // [n][h] output weights (stride 68: conflict-free)
  __shared__ float hid[8][16][68];     // [wave][row][h] tanh(hidden) staging (stride 68)

  const int tid  = threadIdx.x;
  const int wave = tid >> 5;
  const int lane = tid & 31;
  const int m    = lane & 15;          // A-row / C-column index for this lane
  const int hi   = lane >> 4;          // K-half select in A/B operand layouts

  const int b    = blockIdx.x >> 4;    // batch row
  const int grp  = blockIdx.x & 15;
  const int tile = grp * 8 + wave;     // 128 tiles of 16 timesteps per batch
  const int t0   = tile * 16;

  // ---- Stage block-shared weights into LDS (once per block) ----
  for (int idx = tid; idx < 64 * 20; idx += 256) {
    int h = idx / 20, k = idx % 20;
    float v;
    if      (k < 6)  v = W_ar[h * 6 + k];                 // AR lag 0
    else if (k < 12) v = W_ar[384 + h * 6 + (k - 6)];     // AR lag 1
    else if (k < 18) v = W_ma[h * 6 + (k - 12)];          // MA lag 0
    else             v = 0.0f;                            // K padding 18..19
    WaugP[h][k] = v;
  }
  for (int idx = tid; idx < 16 * 64; idx += 256) {
    int n = idx >> 6, h = idx & 63;
    float v = 0.0f;
    if      (n < 3)  v = W_out[n * 64 + h];
    else if (n == 3) v = W_gh[h];
    else if (n == 4) v = W_ga[h];
    Wo2P[n][h] = v;
  }
  __syncthreads();

  // ---- Stage 1 A operand: X_aug[16 x 20] in f32 WMMA A layout ----
  // lane holds row m; VGPR j of chunk kc is X_aug[m][4*kc + 2*hi + j]
  const int t = t0 + m;
  const float* Yrow  = Y + ((size_t)b * Sn + t) * In;
  const float* Yprev = Yrow - In;
  const float* Erow  = E + ((size_t)b * Sn + t) * In;

  auto xval = [&](int k) -> float {
    if (k < 6)  return Yrow[k];
    if (k < 12) return (t > 0) ? Yprev[k - 6] : 0.0f;     // lag-1 missing at t==0
    if (k < 18) return Erow[k - 12];
    return 0.0f;
  };

  v2f A1[5];
#pragma unroll
  for (int kc = 0; kc < 5; ++kc) {
    int k0 = 4 * kc + 2 * hi;
    A1[kc].x = xval(k0);
    A1[kc].y = xval(k0 + 1);
  }

  // ---- Stage 1 WMMAs: 4 N-tiles x 5 K-steps of 16x16x4 fp32 ----
  v8f c[4];
#pragma unroll
  for (int nt = 0; nt < 4; ++nt) {
    const v2f* wrow = (const v2f*)&WaugP[nt * 16 + m][0];  // B column = nt*16 + m
    v8f acc = {};
#pragma unroll
    for (int kc = 0; kc < 5; ++kc) {
      v2f bm = wrow[2 * kc + hi];                          // {W[k0][h], W[k0+1][h]}
      acc = __builtin_amdgcn_wmma_f32_16x16x4_f32(
          /*neg_a=*/false, A1[kc], /*neg_b=*/false, bm,
          /*c_mod=*/(short)0, acc, /*reuse_a=*/false, /*reuse_b=*/false);
    }
    c[nt] = acc;
  }

  // ---- Bias + tanh, re-layout through LDS for stage-2 A operand ----
  // C layout: vgpr v holds element [row = v + 8*hi, col = m] of the 16x16 tile
#pragma unroll
  for (int nt = 0; nt < 4; ++nt) {
    int h = nt * 16 + m;
    float bias1 = b_ar[64 + h];                  // AR lag-1 bias (absent at t==0)
    float bias  = b_ar[h] + bias1 + b_ma[h];
#pragma unroll
    for (int v = 0; v < 8; ++v) {
      int r  = v + 8 * hi;
      int tt = t0 + r;
      float pre = c[nt][v] + bias;
      if (tt == 0) pre -= bias1;
      hid[wave][r][h] = dev_tanh(pre);
    }
  }
  // Same-wave LDS write->read: DS ops are in-order per wave, no barrier needed.

  // ---- Stage 2 WMMAs: [16 x 64] @ [64 x 16], 16 K-steps ----
  v8f c2 = {};
  {
    const v2f* hrow = (const v2f*)&hid[wave][m][0];        // A row = timestep m
    const v2f* orow = (const v2f*)&Wo2P[m][0];             // B column = output n = m
#pragma unroll
    for (int kc = 0; kc < 16; ++kc) {
      v2f a2 = hrow[2 * kc + hi];
      v2f b2 = orow[2 * kc + hi];
      c2 = __builtin_amdgcn_wmma_f32_16x16x4_f32(
          false, a2, false, b2, (short)0, c2, false, false);
    }
  }

  // ---- Scatter outputs: class_logits [B,S,3] | home [B,S] | away [B,S] ----
  {
    const int n = m;                               // output column for this lane
    float ob = 0.0f;
    if      (n < 3)  ob = b_out[n];
    else if (n == 3) ob = b_gh[0];
    else if (n == 4) ob = b_ga[0];
#pragma unroll
    for (int v = 0; v < 8; ++v) {
      int tt = t0 + v + 8 * hi;
      float val = c2[v] + ob;
      size_t bs = (size_t)b * Sn + tt;
      if      (n < 3)  out[bs * 3 + n] = val;
      else if (n == 3) out[(size_t)Bn * Sn * 3 + bs] = val;
      else if (n == 4) out[(size_t)Bn * Sn * 4 + bs] = val;
    }
  }
}

extern "C" void kernel_launch(void* const* d_in, const int* in_sizes, int n_in,
                              void* d_out, int out_size, void* d_ws, size_t ws_size,
                              hipStream_t stream) {
  (void)in_sizes; (void)n_in; (void)out_size; (void)d_ws; (void)ws_size;
  varma_wmma_kernel<<<dim3(Bn * 16), dim3(256), 0, stream>>>(
      (const float*)d_in[0],  (const float*)d_in[1],
      (const float*)d_in[2],  (const float*)d_in[3],
      (const float*)d_in[4],  (const float*)d_in[5],
      (const float*)d_in[6],  (const float*)d_in[7],
      (const float*)d_in[8],  (const float*)d_in[9],
      (const float*)d_in[10], (const float*)d_in[11],
      (float*)d_out);
}